// GraphSAGE_cell_25125558682342
// MI455X (gfx1250) — compile-verified
//
#include <hip/hip_runtime.h>

#define NB 2
#define NN 1024
#define NC 64
#define NH 128

typedef __attribute__((ext_vector_type(2))) float v2f;
typedef __attribute__((ext_vector_type(8))) float v8f;

// Out[m,n] = sum_k In[m,k] * W[k,n] + bias[n], optional ReLU.
// One wave32 per 16x16 output tile; K consumed 4 per V_WMMA_F32_16X16X4_F32.
__global__ __launch_bounds__(256) void wmma_gemm_bias(
    const float* __restrict__ In, int lda,
    const float* __restrict__ W, int ldb,
    const float* __restrict__ bias,
    float* __restrict__ Out, int ldc,
    int M, int Nout, int K, int relu)
{
  const int lane   = threadIdx.x & 31;
  const int wave   = threadIdx.x >> 5;
  const int tilesN = Nout >> 4;
  const int tile   = blockIdx.x * 8 + wave;
  if (tile >= (M >> 4) * tilesN) return;     // wave-uniform (grids are exact)
  const int m0 = (tile / tilesN) << 4;
  const int n0 = (tile % tilesN) << 4;

  const int l16  = lane & 15;
  const int half = lane >> 4;                // lanes 0-15 -> 0, 16-31 -> 1

  // A-frag: lane holds row m0+l16, K = 2*half + {0,1}  (contiguous -> float2)
  const float* arow = In + (size_t)(m0 + l16) * lda + (half << 1);
  // B-frag: lane holds col n0+l16, K = 2*half + {0,1}  (two ldb-strided loads)
  const float* bcol = W + (size_t)(half << 1) * ldb + (n0 + l16);

  v8f acc = {};
  for (int k = 0; k < K; k += 4) {
    v2f a = *(const v2f*)(arow + k);
    v2f b;
    b[0] = bcol[(size_t)k * ldb];
    b[1] = bcol[(size_t)(k + 1) * ldb];
    acc = __builtin_amdgcn_wmma_f32_16x16x4_f32(false, a, false, b,
                                                (short)0, acc, false, false);
  }

  const float bv = bias[n0 + l16];
  // D layout: VGPR r -> row m0 + r + 8*half, col n0 + l16
  float* orow = Out + (size_t)(m0 + (half << 3)) * ldc + (n0 + l16);
#pragma unroll
  for (int r = 0; r < 8; ++r) {
    float v = acc[r] + bv;
    if (relu) v = fmaxf(v, 0.0f);
    orow[(size_t)r * ldc] = v;
  }
}

// message[b,i,c] = max_j A[b,i,j] * Hp[b,j,c]   (A binary, Hp >= 0 -> baseline 0)
// One 256-thread block per destination node: compact nonzero j's into LDS,
// then 4 groups x 64 channels gather only live Hp rows (L2-resident).
__global__ __launch_bounds__(256) void maxagg(
    const float* __restrict__ A, const float* __restrict__ Hp,
    float* __restrict__ Msg)
{
  __shared__ int   s_idx[NN];
  __shared__ int   s_cnt;
  __shared__ float s_part[4][NC];

  const int bi = blockIdx.x;                       // b*NN + i
  const float* Arow = A + (size_t)bi * NN;         // (b*NN+i)*NN == b*NN*NN + i*NN

  if (threadIdx.x == 0) s_cnt = 0;
  __syncthreads();

#pragma unroll
  for (int t = 0; t < NN / 256; ++t) {
    int j = threadIdx.x + t * 256;                 // coalesced pass over A row
    if (Arow[j] != 0.0f) {
      int p = atomicAdd(&s_cnt, 1);
      s_idx[p] = j;
    }
  }
  __syncthreads();

  const int cnt = s_cnt;
  const int g = threadIdx.x >> 6;                  // 4 groups of 64 lanes
  const int c = threadIdx.x & 63;
  const float* HpB = Hp + (size_t)(bi >> 10) * NN * NC;

  float m = 0.0f;                                  // zeros from non-neighbors
  for (int e = g; e < cnt; e += 4) {
    m = fmaxf(m, HpB[(size_t)s_idx[e] * NC + c]);  // 64-ch coalesced gather
  }
  s_part[g][c] = m;
  __syncthreads();
  if (g == 0) {
    m = fmaxf(fmaxf(s_part[0][c], s_part[1][c]),
              fmaxf(s_part[2][c], s_part[3][c]));
    Msg[(size_t)bi * NC + c] = m;
  }
}

extern "C" void kernel_launch(void* const* d_in, const int* in_sizes, int n_in,
                              void* d_out, int out_size, void* d_ws, size_t ws_size,
                              hipStream_t stream)
{
  const float* X  = (const float*)d_in[0];
  const float* A  = (const float*)d_in[1];
  const float* Ws = (const float*)d_in[2];
  const float* Wn = (const float*)d_in[3];
  const float* bs = (const float*)d_in[4];   // bias_self
  const float* bn = (const float*)d_in[5];   // bias_neighbors
  const float* Wp = (const float*)d_in[6];
  const float* bp = (const float*)d_in[7];
  float* out = (float*)d_out;

  float* Hp  = (float*)d_ws;                 // (B*N, C)  = 512 KB
  float* Msg = Hp + (size_t)NB * NN * NC;    // (B*N, C)  = 512 KB

  const int M = NB * NN;                     // 2048 rows

  // 1) Hp = relu(X @ Wpool + bias_pool):  2048x64x64, 512 tiles / 8 waves = 64 blocks
  wmma_gemm_bias<<<(M / 16) * (NC / 16) / 8, 256, 0, stream>>>(
      X, NC, Wp, NC, bp, Hp, NC, M, NC, NC, 1);

  // 2) message = masked row-wise max over neighbors
  maxagg<<<M, 256, 0, stream>>>(A, Hp, Msg);

  // 3) out[:, :128] = message @ Wn + bias_self  (biases swapped as in reference)
  wmma_gemm_bias<<<(M / 16) * (NH / 16) / 8, 256, 0, stream>>>(
      Msg, NC, Wn, NH, bs, out, 2 * NH, M, NH, NC, 0);

  // 4) out[:, 128:] = X @ Ws + bias_neighbors
  wmma_gemm_bias<<<(M / 16) * (NH / 16) / 8, 256, 0, stream>>>(
      X, NC, Ws, NH, bn, out + NH, 2 * NH, M, NH, NC, 0);
}